// neighborhood_routing_algorithm_76501957476790
// MI455X (gfx1250) — compile-verified
//
#include <hip/hip_runtime.h>

typedef __attribute__((ext_vector_type(2))) float v2f;
typedef __attribute__((ext_vector_type(8))) float v8f;

namespace {
constexpr int kD = 10;           // feature dim (K of scores GEMM)
constexpr int kM = 4096;         // softmax axis
constexpr int kN = 16384;        // output rows
constexpr int kSlices = 4;       // s_stack slices
constexpr float kEps = 1e-5f;
constexpr float kShift = 20.0f;  // fixed softmax shift (shift-invariant; range-safe)
constexpr int kWaves = 8;        // waves per block
constexpr int kRowTiles = 2;     // 16-row tiles per wave
constexpr int kTiles = kM / 16;  // 256 m-tiles
constexpr int kS1FragV2 = kTiles * 3 * 32;  // 24576 v2f
constexpr int kSSFragV2 = kTiles * 4 * 32;  // 32768 v2f
}  // namespace

// ---- prepack s_stack[1] fragments: value[(m = 16t + l16)][(d = 4c + 2*(l/16) + {0,1})]
// Serves as the A operand of S^T = s1 @ wrt^T (A layout == B layout on this grid).
__global__ void pack_s1_kernel(const float* __restrict__ s_stack, v2f* __restrict__ frag) {
  int i = blockIdx.x * blockDim.x + threadIdx.x;
  if (i >= kS1FragV2) return;
  const float* s1 = s_stack + (size_t)kM * kD;  // slice i=1 (only last iter survives)
  const int lane = i & 31, c = (i >> 5) % 3, t = i / 96;
  const int row = 16 * t + (lane & 15);
  const int d0 = 4 * c + 2 * (lane >> 4);
  v2f r;
  r.x = (d0 < kD) ? s1[row * kD + d0] : 0.0f;
  r.y = (d0 + 1 < kD) ? s1[row * kD + d0 + 1] : 0.0f;
  frag[i] = r;
}

// ---- prepack s_sum = sum_k s_stack[k] fragments:
// value[(m = 16t + 4c + 2*(l/16) + {0,1})][(d = l16)]; row d == 10 is all-ones so the
// accumulator row 10 carries the softmax denominator. A operand of temp^T = s_sum^T @ P^T.
__global__ void pack_ssum_kernel(const float* __restrict__ s_stack, v2f* __restrict__ frag) {
  int i = blockIdx.x * blockDim.x + threadIdx.x;
  if (i >= kSSFragV2) return;
  const int lane = i & 31, c = (i >> 5) & 3, t = i >> 7;
  const int col = lane & 15;
  const int mr = 16 * t + 4 * c + 2 * (lane >> 4);
  v2f r;
  if (col < kD) {
    float x0 = 0.0f, x1 = 0.0f;
#pragma unroll
    for (int k = 0; k < kSlices; ++k) {
      x0 += s_stack[k * kM * kD + mr * kD + col];
      x1 += s_stack[k * kM * kD + (mr + 1) * kD + col];
    }
    r.x = x0;
    r.y = x1;
  } else {
    const float o = (col == kD) ? 1.0f : 0.0f;  // ones row -> row sums of P
    r.x = o;
    r.y = o;
  }
  frag[i] = r;
}

template <bool PRE>
__device__ __forceinline__ v2f load_s1_frag(const v2f* __restrict__ frag,
                                            const float* __restrict__ s_stack, int t,
                                            int c, int lane) {
  if (PRE) return frag[(t * 3 + c) * 32 + lane];
  const float* s1 = s_stack + (size_t)kM * kD;
  const int row = 16 * t + (lane & 15);
  const int d0 = 4 * c + 2 * (lane >> 4);
  v2f r;
  r.x = (d0 < kD) ? s1[row * kD + d0] : 0.0f;
  r.y = (d0 + 1 < kD) ? s1[row * kD + d0 + 1] : 0.0f;
  return r;
}

template <bool PRE>
__device__ __forceinline__ v2f load_ss_frag(const v2f* __restrict__ frag,
                                            const float* __restrict__ s_stack, int t,
                                            int c, int lane) {
  if (PRE) return frag[(t * 4 + c) * 32 + lane];
  const int col = lane & 15;
  const int mr = 16 * t + 4 * c + 2 * (lane >> 4);
  v2f r;
  if (col < kD) {
    float x0 = 0.0f, x1 = 0.0f;
#pragma unroll
    for (int k = 0; k < kSlices; ++k) {
      x0 += s_stack[k * kM * kD + mr * kD + col];
      x1 += s_stack[k * kM * kD + (mr + 1) * kD + col];
    }
    r.x = x0;
    r.y = x1;
  } else {
    const float o = (col == kD) ? 1.0f : 0.0f;
    r.x = o;
    r.y = o;
  }
  return r;
}

// ---- fused: out = LayerNorm(wrt + softmax(wrt @ s1^T) @ s_sum), all-transposed tiles.
// One wave owns 32 rows (2x16 row tiles); streams 256 m-tiles; no LDS memory, no barriers.
template <bool PRE>
__global__ __launch_bounds__(kWaves * 32) void fused_kernel(
    const float* __restrict__ wrt, const float* __restrict__ s_stack,
    const v2f* __restrict__ fragS1, const v2f* __restrict__ fragSS,
    const float* __restrict__ gamma, const float* __restrict__ beta,
    float* __restrict__ out) {
  const int lane = threadIdx.x & 31;
  const int half = lane >> 4;
  const int l16 = lane & 15;
  const int nbase = (blockIdx.x * kWaves + (threadIdx.x >> 5)) * (16 * kRowTiles);

  // wrt row tiles as B fragments of S^T: (K = d = 4c + 2*half + {0,1}, N = n = l16)
  v2f bW[kRowTiles][3];
#pragma unroll
  for (int r = 0; r < kRowTiles; ++r)
#pragma unroll
    for (int c = 0; c < 3; ++c) {
      const int d0 = 4 * c + 2 * half;
      const int row = nbase + 16 * r + l16;
      bW[r][c].x = (d0 < kD) ? wrt[row * kD + d0] : 0.0f;
      bW[r][c].y = (d0 + 1 < kD) ? wrt[row * kD + d0 + 1] : 0.0f;
    }

  const v8f vzero = {};
  v8f acc[kRowTiles];  // temp^T in D layout: lane l, vgpr v -> (d = v + 8*half, n = l16)
#pragma unroll
  for (int r = 0; r < kRowTiles; ++r) acc[r] = vzero;

  v2f b[3], bs[4];
#pragma unroll
  for (int c = 0; c < 3; ++c) b[c] = load_s1_frag<PRE>(fragS1, s_stack, 0, c, lane);
#pragma unroll
  for (int c = 0; c < 4; ++c) bs[c] = load_ss_frag<PRE>(fragSS, s_stack, 0, c, lane);

  for (int t = 0; t < kTiles; ++t) {
    // software pipeline: issue next tile's fragment loads first
    const int tn = (t + 1 < kTiles) ? t + 1 : t;
    v2f nb[3], nbs[4];
#pragma unroll
    for (int c = 0; c < 3; ++c) nb[c] = load_s1_frag<PRE>(fragS1, s_stack, tn, c, lane);
#pragma unroll
    for (int c = 0; c < 4; ++c) nbs[c] = load_ss_frag<PRE>(fragSS, s_stack, tn, c, lane);

    // S^T = s1_tile(A) @ wrt_tile^T(B): D layout (m = v + 8*half, n = l16)
    v8f sT[kRowTiles];
#pragma unroll
    for (int r = 0; r < kRowTiles; ++r) {
      sT[r] = vzero;
#pragma unroll
      for (int c = 0; c < 3; ++c)
        sT[r] = __builtin_amdgcn_wmma_f32_16x16x4_f32(false, b[c], false, bW[r][c],
                                                      (short)0, sT[r], false, false);
    }

#pragma unroll
    for (int r = 0; r < kRowTiles; ++r) {
      // P^T = exp(S^T - shift) in registers; q = other half's copy (for K spanning halves)
      float p[8], q[8];
#pragma unroll
      for (int v = 0; v < 8; ++v) p[v] = __expf(sT[r][v] - kShift);
#pragma unroll
      for (int v = 0; v < 8; ++v) q[v] = __shfl_xor(p[v], 16, 32);

      // acc += s_sum_tile^T(A) @ P^T(B); B-frag (K = 4c + 2*half + {0,1}, N = l16):
      //   K-rows 0..7 live in half 0 lanes, 8..15 in half 1 lanes -> own/partner select.
#pragma unroll
      for (int c = 0; c < 4; ++c) {
        const int base = 4 * (c & 1);
        v2f b2;
        if (c < 2) {  // source K-rows 0..7 (half 0 registers)
          b2.x = half ? q[base + 2] : p[base];
          b2.y = half ? q[base + 3] : p[base + 1];
        } else {      // source K-rows 8..15 (half 1 registers)
          b2.x = half ? p[base + 2] : q[base];
          b2.y = half ? p[base + 3] : q[base + 1];
        }
        acc[r] = __builtin_amdgcn_wmma_f32_16x16x4_f32(false, bs[c], false, b2, (short)0,
                                                       acc[r], false, false);
      }
    }

#pragma unroll
    for (int c = 0; c < 3; ++c) b[c] = nb[c];
#pragma unroll
    for (int c = 0; c < 4; ++c) bs[c] = nbs[c];
  }

  // ---- epilogue: temp = acc[d]/acc[10]; out = LayerNorm(wrt + temp). d in registers.
  float gv[8], bv[8];
#pragma unroll
  for (int v = 0; v < 8; ++v) {
    const int d = v + 8 * half;
    const int sd = (d < kD) ? d : 0;
    gv[v] = gamma[sd];
    bv[v] = beta[sd];
  }
#pragma unroll
  for (int r = 0; r < kRowTiles; ++r) {
    const int n = nbase + 16 * r + l16;
    const float sw2 = __shfl_xor(acc[r][2], 16, 32);       // row d=10 lives in half 1, v=2
    const float lsum = half ? acc[r][2] : sw2;
    const float rinv = 1.0f / lsum;
    float x[8], S = 0.0f;
#pragma unroll
    for (int v = 0; v < 8; ++v) {
      const int d = v + 8 * half;
      const int sd = (d < kD) ? d : 0;
      const float w = wrt[n * kD + sd];
      x[v] = (d < kD) ? (w + acc[r][v] * rinv) : 0.0f;
      S += x[v];
    }
    const float mu = (S + __shfl_xor(S, 16, 32)) * (1.0f / kD);
    float dv[8], SS = 0.0f;
#pragma unroll
    for (int v = 0; v < 8; ++v) {
      const int d = v + 8 * half;
      dv[v] = (d < kD) ? (x[v] - mu) : 0.0f;
      SS += dv[v] * dv[v];
    }
    const float var = (SS + __shfl_xor(SS, 16, 32)) * (1.0f / kD);
    const float rstd = rsqrtf(var + kEps);
#pragma unroll
    for (int v = 0; v < 8; ++v) {
      const int d = v + 8 * half;
      if (d < kD) out[n * kD + d] = dv[v] * rstd * gv[v] + bv[v];
    }
  }
}

extern "C" void kernel_launch(void* const* d_in, const int* in_sizes, int n_in,
                              void* d_out, int out_size, void* d_ws, size_t ws_size,
                              hipStream_t stream) {
  const float* s_stack = (const float*)d_in[0];  // [4, 4096, 10]
  const float* wrt = (const float*)d_in[1];      // [16384, 10]
  const float* gamma = (const float*)d_in[2];    // [10]
  const float* beta = (const float*)d_in[3];     // [10]
  float* out = (float*)d_out;                    // [16384, 10]

  const size_t frag_bytes = (size_t)(kS1FragV2 + kSSFragV2) * sizeof(v2f);  // 448 KB
  const int blocks = kN / (16 * kRowTiles * kWaves);  // 64 blocks x 256 threads

  if (ws_size >= frag_bytes) {
    v2f* fragS1 = (v2f*)d_ws;
    v2f* fragSS = fragS1 + kS1FragV2;
    pack_s1_kernel<<<(kS1FragV2 + 255) / 256, 256, 0, stream>>>(s_stack, fragS1);
    pack_ssum_kernel<<<(kSSFragV2 + 255) / 256, 256, 0, stream>>>(s_stack, fragSS);
    fused_kernel<true><<<blocks, kWaves * 32, 0, stream>>>(wrt, s_stack, fragS1, fragSS,
                                                           gamma, beta, out);
  } else {
    fused_kernel<false><<<blocks, kWaves * 32, 0, stream>>>(wrt, s_stack, nullptr,
                                                            nullptr, gamma, beta, out);
  }
}